// MXIntGELU_27779848471232
// MI455X (gfx1250) — compile-verified
//
#include <hip/hip_runtime.h>

// MXIntGELU on MI455X (gfx1250, wave32).
//
// Reference simplifies to: out = mxint_quant(relu(x)), block size 32 along
// the contiguous dim, width=8 (6 fractional bits), exp_width=8.
//
// Bandwidth-bound: 512 MB HBM traffic -> ~22 us floor at 23.3 TB/s.
// Design:
//  - each lane owns 8 contiguous floats (2x 128-bit NT loads/stores)
//  - one 32-elem quant block == 4 consecutive lanes -> 2x ds_swizzle XOR max
//  - non-temporal hints so the 512 MB stream doesn't churn the 192 MB L2
//  - exponent via IEEE bit extraction; scaling via v_ldexp_f32 (exact, and
//    safe for e=-126 where a float inverse scale 2^132 would overflow)
//  - round-half-to-even via v_rndne_f32 to match jnp.round

typedef float v4f __attribute__((ext_vector_type(4)));  // native clang vector

#define ELEMS_PER_LANE 8   // 2 x v4f
#define BLOCK_THREADS  256

// ds_swizzle group-of-32 encoding: offset[14:10]=xor, [9:5]=or, [4:0]=and
// SWAPX1 = (1<<10)|0x1f = 0x041F ; SWAPX2 = (2<<10)|0x1f = 0x081F
__device__ __forceinline__ float swz_max_x1(float v) {
    int i = __builtin_amdgcn_ds_swizzle(__float_as_int(v), 0x041F);
    return fmaxf(v, __int_as_float(i));
}
__device__ __forceinline__ float swz_max_x2(float v) {
    int i = __builtin_amdgcn_ds_swizzle(__float_as_int(v), 0x081F);
    return fmaxf(v, __int_as_float(i));
}

__device__ __forceinline__ float quant_one(float q, int k_fwd, int k_bwd) {
    // q >= 0 (post-ReLU). m = rne(q * 2^k_fwd), clamp <=127, out = m * 2^k_bwd
    float m = rintf(ldexpf(q, k_fwd));
    m = fminf(m, 127.0f);
    return ldexpf(m, k_bwd);
}

__global__ __launch_bounds__(BLOCK_THREADS) void mxint_gelu_kernel(
    const float* __restrict__ x, float* __restrict__ out, long long n) {
    const long long tid  = (long long)blockIdx.x * BLOCK_THREADS + threadIdx.x;
    const long long base = tid * ELEMS_PER_LANE;
    if (base + ELEMS_PER_LANE > n) return;   // n is a multiple of 32; whole
                                             // 4-lane groups retire together

    const v4f* __restrict__ xin = reinterpret_cast<const v4f*>(x + base);
    v4f a = __builtin_nontemporal_load(xin);
    v4f b = __builtin_nontemporal_load(xin + 1);

    // ReLU (qout = x * 0.5 * (1 + sign(x)))
    a.x = fmaxf(a.x, 0.0f); a.y = fmaxf(a.y, 0.0f);
    a.z = fmaxf(a.z, 0.0f); a.w = fmaxf(a.w, 0.0f);
    b.x = fmaxf(b.x, 0.0f); b.y = fmaxf(b.y, 0.0f);
    b.z = fmaxf(b.z, 0.0f); b.w = fmaxf(b.w, 0.0f);

    // Block max over 32 contiguous elements = this lane's 8 + 4-lane group
    float lm = fmaxf(fmaxf(fmaxf(a.x, a.y), fmaxf(a.z, a.w)),
                     fmaxf(fmaxf(b.x, b.y), fmaxf(b.z, b.w)));
    lm = swz_max_x1(lm);   // xor 1 within group of 4
    lm = swz_max_x2(lm);   // xor 2 within group of 4

    // e = floor(log2(max(|q|, 2^-126))) via exponent-field extraction
    float ma = fmaxf(lm, 1.17549435e-38f);             // 2^-126, forces normal
    int   e  = (int)(__float_as_uint(ma) >> 23) - 127; // sign bit is 0
    // e in [-126, 127] -> already inside the exp_width=8 clip range
    const int k_fwd = 6 - e;   // q / 2^(e-6)  == ldexp(q, 6-e), exact
    const int k_bwd = e - 6;   // m * 2^(e-6)  == ldexp(m, e-6), exact

    v4f oa, ob;
    oa.x = quant_one(a.x, k_fwd, k_bwd); oa.y = quant_one(a.y, k_fwd, k_bwd);
    oa.z = quant_one(a.z, k_fwd, k_bwd); oa.w = quant_one(a.w, k_fwd, k_bwd);
    ob.x = quant_one(b.x, k_fwd, k_bwd); ob.y = quant_one(b.y, k_fwd, k_bwd);
    ob.z = quant_one(b.z, k_fwd, k_bwd); ob.w = quant_one(b.w, k_fwd, k_bwd);

    v4f* __restrict__ oout = reinterpret_cast<v4f*>(out + base);
    __builtin_nontemporal_store(oa, oout);
    __builtin_nontemporal_store(ob, oout + 1);
}

extern "C" void kernel_launch(void* const* d_in, const int* in_sizes, int n_in,
                              void* d_out, int out_size, void* d_ws, size_t ws_size,
                              hipStream_t stream) {
    (void)n_in; (void)d_ws; (void)ws_size; (void)out_size;
    const float* x  = (const float*)d_in[0];
    float*      out = (float*)d_out;
    const long long n = (long long)in_sizes[0];   // 4*4096*4096 = 64Mi

    const long long threads = (n + ELEMS_PER_LANE - 1) / ELEMS_PER_LANE;
    const long long blocks  = (threads + BLOCK_THREADS - 1) / BLOCK_THREADS;
    mxint_gelu_kernel<<<dim3((unsigned)blocks), dim3(BLOCK_THREADS), 0, stream>>>(x, out, n);
}